// CombineRadialSpeciesWithAngularAdaptBasis_48060684042914
// MI455X (gfx1250) — compile-verified
//
#include <hip/hip_runtime.h>

typedef __attribute__((ext_vector_type(2))) float v2f;
typedef __attribute__((ext_vector_type(4))) float v4f;
typedef __attribute__((ext_vector_type(8))) float v8f;

#define PDIM 80          // K dimension
#define BDIM 64          // output columns
#define KSTEPS 20        // 80 / 4
#define WAVES_PER_BLOCK 8
#define ROWS_PER_WAVE 16
#define ROWS_PER_BLOCK (WAVES_PER_BLOCK * ROWS_PER_WAVE)  // 128
#define AROW_STRIDE 84   // padded LDS row stride (floats) -> conflict-free ds_load_b64

struct SegDesc {
  const float* A;   // [rows, 80] row-major
  const float* W;   // [80, 64] row-major
  float* O;         // [rows, 64] row-major
  int rows;
  int tileStart;    // prefix sum of block tiles
};
struct Params { SegDesc seg[12]; };

__global__ __launch_bounds__(256) void
combine_wmma_kernel(Params p) {
  // W swizzled for WMMA-B operand: lW[ks][half][n][j] = W[ks*4 + half*2 + j][n]
  __shared__ float lW[KSTEPS * 2 * BDIM * 2];                       // 5120 floats = 20 KB
  __shared__ float lA[WAVES_PER_BLOCK * ROWS_PER_WAVE * AROW_STRIDE]; // 10752 floats = 42 KB

  const int tid  = threadIdx.x;
  const int bid  = blockIdx.x;

  // Map block -> segment (12 entries, unrolled scan)
  int s = 0;
#pragma unroll
  for (int i = 1; i < 12; ++i)
    if (bid >= p.seg[i].tileStart) s = i;
  const SegDesc sd = p.seg[s];
  const int rowBlockBase = (bid - sd.tileStart) * ROWS_PER_BLOCK;

  // ---- Stage W into LDS in WMMA-B layout (cooperative, 20 elems/thread) ----
  {
    const float* __restrict__ W = sd.W;
#pragma unroll
    for (int t = 0; t < (KSTEPS * 2 * BDIM * 2) / 256; ++t) {
      int idx = tid + t * 256;
      int j   = idx & 1;
      int n   = (idx >> 1) & 63;
      int kh  = idx >> 7;          // kh = ks*2 + half, 0..39
      lW[idx] = W[(kh * 2 + j) * BDIM + n];
    }
  }
  __syncthreads();

  const int wave     = tid >> 5;
  const int lane     = tid & 31;
  const int laneHalf = lane >> 4;   // 0: lanes 0-15, 1: lanes 16-31
  const int lane16   = lane & 15;

  const int rowBase = rowBlockBase + wave * ROWS_PER_WAVE;
  if (rowBase >= sd.rows) return;   // wave-uniform; all row counts are multiples of 16

  // ---- Stage this wave's 16x80 A tile into LDS (coalesced float4 loads) ----
  float* lAw = &lA[wave * ROWS_PER_WAVE * AROW_STRIDE];
  {
    const float* __restrict__ Ag = sd.A + (size_t)rowBase * PDIM;
#pragma unroll
    for (int i = 0; i < 10; ++i) {        // 16*80/4 = 320 float4, 32 lanes -> 10 each
      int f4   = lane + i * 32;
      int flat = f4 * 4;
      int r    = flat / PDIM;
      int c    = flat % PDIM;             // multiple of 4
      v4f v = __builtin_nontemporal_load((const v4f*)(Ag + flat));
      *(v4f*)(lAw + r * AROW_STRIDE + c) = v;
    }
  }
  // Same-wave LDS accesses are in-order (DS pipe) -> safe to read back immediately.

  // ---- 16x64 tile: 20 k-steps x 4 n-tiles of v_wmma_f32_16x16x4_f32 ----
  v8f acc0 = {}, acc1 = {}, acc2 = {}, acc3 = {};
  const float* aBase = lAw + lane16 * AROW_STRIDE + laneHalf * 2;
  const float* bBase = lW + laneHalf * 128 + lane16 * 2;

#pragma unroll
  for (int ks = 0; ks < KSTEPS; ++ks) {
    v2f a  = *(const v2f*)(aBase + ks * 4);
    v2f b0 = *(const v2f*)(bBase + ks * 256 + 0 * 32);
    v2f b1 = *(const v2f*)(bBase + ks * 256 + 1 * 32);
    v2f b2 = *(const v2f*)(bBase + ks * 256 + 2 * 32);
    v2f b3 = *(const v2f*)(bBase + ks * 256 + 3 * 32);
    acc0 = __builtin_amdgcn_wmma_f32_16x16x4_f32(false, a, false, b0, (short)0, acc0, false, false);
    acc1 = __builtin_amdgcn_wmma_f32_16x16x4_f32(false, a, false, b1, (short)0, acc1, false, false);
    acc2 = __builtin_amdgcn_wmma_f32_16x16x4_f32(false, a, false, b2, (short)0, acc2, false, false);
    acc3 = __builtin_amdgcn_wmma_f32_16x16x4_f32(false, a, false, b3, (short)0, acc3, false, false);
  }

  // ---- Store C: VGPR r -> row (r + 8*laneHalf), col lane16 (+16 per n-tile) ----
  float* __restrict__ Og = sd.O + (size_t)rowBase * BDIM;
  const int rOff = laneHalf * 8;
#pragma unroll
  for (int r = 0; r < 8; ++r) {
    float* orow = Og + (size_t)(r + rOff) * BDIM + lane16;
    __builtin_nontemporal_store(acc0[r], orow + 0);
    __builtin_nontemporal_store(acc1[r], orow + 16);
    __builtin_nontemporal_store(acc2[r], orow + 32);
    __builtin_nontemporal_store(acc3[r], orow + 48);
  }
}

extern "C" void kernel_launch(void* const* d_in, const int* in_sizes, int n_in,
                              void* d_out, int out_size, void* d_ws, size_t ws_size,
                              hipStream_t stream) {
  (void)in_sizes; (void)n_in; (void)out_size; (void)d_ws; (void)ws_size;
  const int N = 30000, NG = 8000;
  Params p{};
  long long outRows = 0;
  int tiles = 0;
  for (int l = 0; l < 6; ++l) {
    const int m = 2 * l + 1;
    const float* V = (const float*)d_in[3 * l + 0];   // values_l{l}: [N, m, P]
    const float* G = (const float*)d_in[3 * l + 1];   // grads_l{l}:  [NG, 3, m, P]
    const float* W = (const float*)d_in[3 * l + 2];   // W_l{l}:      [P, B]

    const int rv = N * m;          // multiple of 16
    SegDesc& s0 = p.seg[2 * l];
    s0.A = V; s0.W = W; s0.O = (float*)d_out + outRows * BDIM;
    s0.rows = rv; s0.tileStart = tiles;
    tiles += (rv + ROWS_PER_BLOCK - 1) / ROWS_PER_BLOCK;
    outRows += rv;

    const int rg = NG * 3 * m;     // multiple of 16
    SegDesc& s1 = p.seg[2 * l + 1];
    s1.A = G; s1.W = W; s1.O = (float*)d_out + outRows * BDIM;
    s1.rows = rg; s1.tileStart = tiles;
    tiles += (rg + ROWS_PER_BLOCK - 1) / ROWS_PER_BLOCK;
    outRows += rg;
  }
  combine_wmma_kernel<<<dim3(tiles), dim3(256), 0, stream>>>(p);
}